// ModelSglang_28681791602770
// MI455X (gfx1250) — compile-verified
//
#include <hip/hip_runtime.h>
#include <math.h>

// ---------------------------------------------------------------------------
// Merge split-KV attention partials (SGLang decode merge) for MI455X/gfx1250.
//   out[b,h,d] = sum_s w[b,h,s]*att_out[b,h,s,d] / sum_s w[b,h,s]
//   w = exp(lse - max_masked(lse)), mask: s*ceil(seq/nsplit) < seq
// HBM-bandwidth bound: 64MB read + 4MB write @ 23.3 TB/s ~ 3us floor.
// Data path: GLOBAL_LOAD_ASYNC_TO_LDS_B128 streams each wave's 16x128 f32
// tile as 512B-contiguous rows into (bank-conflict-padded) LDS; the 1x16 x
// 16x128 weighted sum runs on chained V_WMMA_F32_16X16X4_F32 with A = X^T
// fragments from LDS and B = merge weights broadcast along N.
// ---------------------------------------------------------------------------

typedef __attribute__((ext_vector_type(2))) float v2f;
typedef __attribute__((ext_vector_type(4))) float v4f;
typedef __attribute__((ext_vector_type(8))) float v8f;

#define BB 256
#define HH 32
#define SS 16
#define DD 128
#define ROWP 144   // padded LDS row stride in dwords: +288 dwords between
                   // half-wave accesses => +32 banks => conflict-free
#define WAVES_PER_BLOCK 8

__global__ __launch_bounds__(32 * WAVES_PER_BLOCK)
void merge_kv_splits_kernel(const float* __restrict__ att_out,
                            const float* __restrict__ att_lse,
                            const int*   __restrict__ kv_indptr,
                            const int*   __restrict__ num_kv_splits,
                            float*       __restrict__ out)
{
    __shared__ __align__(16) float tile[WAVES_PER_BLOCK * SS * ROWP];
    __shared__ __align__(16) float smem[WAVES_PER_BLOCK * DD];

    const int lane = threadIdx.x & 31;
    const int wave = threadIdx.x >> 5;
    const int idx  = blockIdx.x * WAVES_PER_BLOCK + wave;   // idx = b*H + h
    const int b    = idx / HH;

    // -------- async global->LDS: stream this wave's 16x128 f32 tile --------
    // One instruction per s-row: 32 lanes x 16B = 512B contiguous from HBM.
    // LDS rows are padded to 576B (ROWP dwords) to kill bank conflicts.
    float* tbase = &tile[wave * (SS * ROWP)];
    {
        unsigned lds_addr = (unsigned)(uintptr_t)tbase + (unsigned)(lane * 16);
        unsigned long long gaddr = (unsigned long long)(uintptr_t)att_out
                                 + (unsigned long long)idx * (SS * DD * 4)
                                 + (unsigned long long)(lane * 16);
        #pragma unroll
        for (int r = 0; r < SS; ++r) {
            asm volatile("global_load_async_to_lds_b128 %0, %1, off"
                         :: "v"(lds_addr), "v"(gaddr)
                         : "memory");
            lds_addr += ROWP * 4;   // 576B LDS row stride
            gaddr    += DD * 4;     // 512B global row stride
        }
    }

    // -------- per-batch split geometry (overlaps with async DMA) --------
    const int seq = kv_indptr[b + 1] - kv_indptr[b];
    int ns = num_kv_splits[b];
    ns = ns < 1 ? 1 : (ns > SS ? SS : ns);
    const int kv_per = (seq + ns - 1) / ns;

    const int s   = lane & 15;      // split index this lane owns (dup on 16-31)
    const int hi  = lane >> 4;      // half-wave select
    const int hi2 = hi << 1;        // K offset for A/B layout (0 or 2)

    // -------- masked max / exp / sum over the 16 splits (wave32 shfl) ------
    const float lse  = att_lse[idx * SS + s];
    const bool  live = (s * kv_per) < seq;

    float mval = live ? lse : -INFINITY;
    #pragma unroll
    for (int off = 8; off > 0; off >>= 1)
        mval = fmaxf(mval, __shfl_xor(mval, off, 32));

    const float w = live ? __expf(lse - mval) : 0.0f;

    float den = w;
    #pragma unroll
    for (int off = 8; off > 0; off >>= 1)
        den += __shfl_xor(den, off, 32);
    const float rden = 1.0f / den;

    // -------- B operands: vB[j] covers s = 4j..4j+3, broadcast over N ------
    // B 4x16 f32 layout: VGPR0 = rows K=0 (lanes0-15) / K=2 (lanes16-31),
    //                    VGPR1 = rows K=1 / K=3.
    v2f vB[4];
    #pragma unroll
    for (int j = 0; j < 4; ++j) {
        vB[j].x = __shfl(w, 4 * j + hi2,     32);
        vB[j].y = __shfl(w, 4 * j + hi2 + 1, 32);
    }

    // -------- wait for the tile DMA, then WMMA from LDS --------
    asm volatile("s_wait_asynccnt 0x0" ::: "memory");

    #pragma unroll
    for (int dc = 0; dc < DD / 16; ++dc) {
        const int d0 = dc * 16;
        v8f acc = {};
        #pragma unroll
        for (int j = 0; j < 4; ++j) {
            const int sb = 4 * j + hi2;    // this lane's K rows: sb, sb+1
            v2f a;
            // A 16x4 f32 layout: lane L<16 holds (M=L, K=0/1) in v0/v1,
            //                    lane L>=16 holds (M=L-16, K=2/3).
            a.x = tbase[(sb    ) * ROWP + d0 + s];
            a.y = tbase[(sb + 1) * ROWP + d0 + s];
            acc = __builtin_amdgcn_wmma_f32_16x16x4_f32(
                      false, a, false, vB[j], (short)0, acc, false, false);
        }
        // Every column of D equals y[d0+m]. Lane 0 holds m=0..7 in acc[0..7],
        // lane 16 holds m=8..15. Stage to LDS for a coalesced final store.
        if ((lane & 15) == 0) {
            float* dst = &smem[wave * DD + d0 + hi * 8];
            #pragma unroll
            for (int v = 0; v < 8; ++v) dst[v] = acc[v];
        }
    }

    __syncthreads();

    // -------- normalize + coalesced 512B store per wave --------
    const v4f y = *(const v4f*)&smem[wave * DD + lane * 4];
    v4f o;
    o.x = y.x * rden; o.y = y.y * rden; o.z = y.z * rden; o.w = y.w * rden;
    *(v4f*)&out[(size_t)idx * DD + lane * 4] = o;
}

extern "C" void kernel_launch(void* const* d_in, const int* in_sizes, int n_in,
                              void* d_out, int out_size, void* d_ws, size_t ws_size,
                              hipStream_t stream) {
    // setup_inputs order: att_out, att_lse, q, v_buffer, kv_indptr, num_kv_splits
    const float* att_out       = (const float*)d_in[0];
    const float* att_lse       = (const float*)d_in[1];
    const int*   kv_indptr     = (const int*)d_in[4];
    const int*   num_kv_splits = (const int*)d_in[5];
    float*       out           = (float*)d_out;

    dim3 grid((BB * HH) / WAVES_PER_BLOCK);
    dim3 block(32 * WAVES_PER_BLOCK);
    merge_kv_splits_kernel<<<grid, block, 0, stream>>>(
        att_out, att_lse, kv_indptr, num_kv_splits, out);
}